// CRF_11879879542805
// MI455X (gfx1250) — compile-verified
//
#include <hip/hip_runtime.h>
#include <hip/hip_bf16.h>

#define Bn 2048
#define Tn 512
#define Kn 32

typedef __attribute__((ext_vector_type(2))) float v2f;
typedef __attribute__((ext_vector_type(8))) float v8f;

// ---------------------------------------------------------------------------
// Kernel 1: log_norm via linear-domain forward recurrence on WMMA.
//   a'[b][j] = exp(pot[b][t][j]) * sum_i a[b][i] * E[i][j],  E = exp(trans)
//   per-row running log-scale "logs"; renormalize every 8 steps (f32 range).
// One wave per 16-batch tile: D[16x32] = A[16x32] x E[32x32] via
// 8 K-chunks x 2 N-tiles of V_WMMA_F32_16X16X4_F32.
// ---------------------------------------------------------------------------
__global__ __launch_bounds__(32) void crf_fwd_lognorm(
    const float* __restrict__ pot, const float* __restrict__ trans,
    const int* __restrict__ slen, float* __restrict__ log_norm) {
  __shared__ __align__(16) float tile[16 * Kn];  // 2KB row-major a-tile

  const int lane = threadIdx.x;
  const int ln = lane & 15;   // column-within-tile / A-row
  const int hi = lane >> 4;   // lane half
  const int b0 = blockIdx.x * 16;

  // --- constant B fragments: E = exp(trans), built once per wave ---
  // chunk c covers K rows 4c..4c+3; lo-lanes hold rows 4c,4c+1; hi-lanes 4c+2,4c+3
  v2f Bfr[2][8];
#pragma unroll
  for (int n = 0; n < 2; ++n) {
#pragma unroll
    for (int c = 0; c < 8; ++c) {
      const int k = 4 * c + 2 * hi;
      const int col = n * 16 + ln;
      Bfr[n][c].x = __expf(trans[k * Kn + col]);
      Bfr[n][c].y = __expf(trans[(k + 1) * Kn + col]);
    }
  }

  // --- per-row state in C-layout: VGPR r holds rows {r (lanes0-15), r+8 (16-31)} ---
  float aC[2][8];
  float logs[8];
  int len8[8];
#pragma unroll
  for (int r = 0; r < 8; ++r) {
    const int b = b0 + r + 8 * hi;
    len8[r] = slen[b];
    logs[r] = 0.0f;
    const float* p0 = pot + (size_t)b * Tn * Kn;
    aC[0][r] = __expf(p0[ln]);
    aC[1][r] = __expf(p0[16 + ln]);
  }

  for (int t = 1; t < Tn; ++t) {
    // --- issue this step's 16 pot loads FIRST (latency hidden under LDS+WMMA);
    //     fully unconditional so the compiler cannot sink them into branches ---
    float P0[8], P1[8];
#pragma unroll
    for (int r = 0; r < 8; ++r) {
      const float* pr = pot + ((size_t)(b0 + r + 8 * hi) * Tn + t) * Kn;
      P0[r] = pr[ln];
      P1[r] = pr[16 + ln];
    }

    if ((t & 7) == 1) {
      // periodic per-row renormalization (keeps f32 in range; logs compensates)
#pragma unroll
      for (int r = 0; r < 8; ++r) {
        float m = fmaxf(aC[0][r], aC[1][r]);
#pragma unroll
        for (int off = 1; off < 16; off <<= 1)
          m = fmaxf(m, __shfl_xor(m, off, 32));  // stays within each 16-lane half
        const float inv = 1.0f / m;
        aC[0][r] *= inv;
        aC[1][r] *= inv;
        logs[r] += __logf(m);
      }
    }

    // C-layout -> LDS row-major tile
#pragma unroll
    for (int r = 0; r < 8; ++r) {
      tile[(r + 8 * hi) * Kn + ln] = aC[0][r];
      tile[(r + 8 * hi) * Kn + 16 + ln] = aC[1][r];
    }
    __syncthreads();  // single-wave WG: lowers to ds-count wait only

    // LDS -> A fragments (lane = row ln; K chunk split across lane halves)
    v2f Af[8];
#pragma unroll
    for (int c = 0; c < 8; ++c) {
      Af[c] = *(const v2f*)&tile[ln * Kn + 4 * c + 2 * hi];  // 8B-aligned ds_load_b64
    }

    // D = A x E : 8 chained K=4 WMMAs per 16-col tile
    v8f acc0 = {0.f, 0.f, 0.f, 0.f, 0.f, 0.f, 0.f, 0.f};
    v8f acc1 = {0.f, 0.f, 0.f, 0.f, 0.f, 0.f, 0.f, 0.f};
#pragma unroll
    for (int c = 0; c < 8; ++c) {
      acc0 = __builtin_amdgcn_wmma_f32_16x16x4_f32(false, Af[c], false, Bfr[0][c],
                                                   (short)0, acc0, false, false);
      acc1 = __builtin_amdgcn_wmma_f32_16x16x4_f32(false, Af[c], false, Bfr[1][c],
                                                   (short)0, acc1, false, false);
    }

    // a' = exp(pot_t) * D, blended branchlessly by the sequence-length mask:
    //   msk = 1 -> take new value exactly; msk = 0 -> keep frozen aC exactly.
#pragma unroll
    for (int r = 0; r < 8; ++r) {
      const float msk = (t < len8[r]) ? 1.0f : 0.0f;  // v_cmp + v_cndmask
      const float keep = 1.0f - msk;
      const float n0 = __expf(P0[r]) * acc0[r];
      const float n1 = __expf(P1[r]) * acc1[r];
      aC[0][r] = fmaf(n0, msk, aC[0][r] * keep);
      aC[1][r] = fmaf(n1, msk, aC[1][r] * keep);
    }
    __syncthreads();  // tile reused next iteration
  }

  // log_norm[b] = logs + log(sum_j a[b][j])
#pragma unroll
  for (int r = 0; r < 8; ++r) {
    float s = aC[0][r] + aC[1][r];
#pragma unroll
    for (int off = 1; off < 16; off <<= 1) s += __shfl_xor(s, off, 32);
    if (ln == 0) log_norm[b0 + r + 8 * hi] = logs[r] + __logf(s);
  }
}

// ---------------------------------------------------------------------------
// Kernel 2: Viterbi (max-plus: no WMMA semiring) + in-LDS backpointers +
// in-kernel backtrack. One wave per batch, lane = current state j.
// alpha[i] broadcast via v_readlane_b32 (scalar operand into v_add_f32);
// trans column j resident in 32 VGPRs.
// ---------------------------------------------------------------------------
__global__ __launch_bounds__(32) void crf_viterbi(
    const float* __restrict__ pot, const float* __restrict__ trans,
    const int* __restrict__ slen, float* __restrict__ out_tags,
    float* __restrict__ out_best) {
  __shared__ unsigned char bp[Tn][Kn];  // 16KB backpointers (index 1..T-1)
  __shared__ unsigned char tags[Tn];    // decoded tags

  const int b = blockIdx.x;
  const int j = threadIdx.x;  // current state

  float tcol[Kn];  // trans[:, j]
#pragma unroll
  for (int i = 0; i < Kn; ++i) tcol[i] = trans[i * Kn + j];

  const int len = slen[b];
  const float* pb = pot + (size_t)b * Tn * Kn;
  float alpha = pb[j];  // t = 0

  for (int t = 1; t < Tn; ++t) {
    __builtin_prefetch(pb + (t + 8) * Kn + j, 0, 0);  // global_prefetch_b8
    const int abits = __float_as_int(alpha);
    float best = -3.4e38f;
    int arg = 0;
#pragma unroll
    for (int i = 0; i < Kn; ++i) {
      const float ai =
          __int_as_float(__builtin_amdgcn_readlane(abits, i));  // SGPR broadcast
      const float sc = ai + tcol[i];  // v_add_f32 v, s, v
      const bool gt = sc > best;      // strict > keeps first max (jnp.argmax)
      arg = gt ? i : arg;
      best = gt ? sc : best;
    }
    const bool act = t < len;
    const float na = pb[t * Kn + j] + best;
    alpha = act ? na : alpha;
    bp[t][j] = (unsigned char)(act ? arg : j);  // identity when masked
  }

  // final max/argmax across lanes (first-max tie rule)
  float bv = alpha;
  int bt = j;
#pragma unroll
  for (int off = 16; off > 0; off >>= 1) {
    const float ov = __shfl_down(bv, off, 32);
    const int ot = __shfl_down(bt, off, 32);
    const bool take = (ov > bv) || (ov == bv && ot < bt);
    bv = take ? ov : bv;
    bt = take ? ot : bt;
  }
  if (j == 0) {
    out_best[b] = bv;
    int cur = bt;
    tags[Tn - 1] = (unsigned char)cur;
    for (int t = Tn - 2; t >= 0; --t) {
      cur = bp[t + 1][cur];
      tags[t] = (unsigned char)cur;
    }
  }
  __syncthreads();
  // coalesced tag writeout (output dtype: float)
  for (int t = j; t < Tn; t += 32) out_tags[(size_t)b * Tn + t] = (float)tags[t];
}

// ---------------------------------------------------------------------------
// Kernel 3: sequence score + log_likelihood. One wave per batch.
// ---------------------------------------------------------------------------
__global__ __launch_bounds__(256) void crf_score_ll(
    const float* __restrict__ pot, const float* __restrict__ trans,
    const int* __restrict__ slen, const int* __restrict__ tag_idx,
    const float* __restrict__ log_norm, float* __restrict__ out_ll) {
  const int wid = (blockIdx.x * blockDim.x + threadIdx.x) >> 5;  // uniform per wave
  if (wid >= Bn) return;
  const int lane = threadIdx.x & 31;
  const int b = wid;
  const int len = slen[b];
  const float* pb = pot + (size_t)b * Tn * Kn;
  const int* tb = tag_idx + (size_t)b * Tn;

  float acc = 0.0f;
  for (int t = lane; t < Tn; t += 32) {
    const int tg = tb[t];
    if (t < len) acc += pb[t * Kn + tg];
    if (t + 1 < Tn && t + 1 < len) acc += trans[tg * Kn + tb[t + 1]];
  }
#pragma unroll
  for (int off = 16; off > 0; off >>= 1) acc += __shfl_xor(acc, off, 32);
  if (lane == 0) out_ll[b] = acc - log_norm[b];
}

// ---------------------------------------------------------------------------
extern "C" void kernel_launch(void* const* d_in, const int* in_sizes, int n_in,
                              void* d_out, int out_size, void* d_ws,
                              size_t ws_size, hipStream_t stream) {
  const float* pot = (const float*)d_in[0];     // [B,T,K]
  const float* trans = (const float*)d_in[1];   // [K,K]
  const int* slen = (const int*)d_in[2];        // [B]
  const int* tag_idx = (const int*)d_in[3];     // [B,T]

  float* out = (float*)d_out;
  float* out_tags = out;                          // [B*T] decode_tags (as float)
  float* out_best = out + (size_t)Bn * Tn;        // [B]   best_score
  float* out_ll = out + (size_t)Bn * Tn + Bn;     // [B]   log_likelihood
  float* ws_lognorm = (float*)d_ws;               // [B]   scratch

  crf_fwd_lognorm<<<Bn / 16, 32, 0, stream>>>(pot, trans, slen, ws_lognorm);
  crf_viterbi<<<Bn, 32, 0, stream>>>(pot, trans, slen, out_tags, out_best);
  crf_score_ll<<<(Bn * 32 + 255) / 256, 256, 0, stream>>>(pot, trans, slen,
                                                          tag_idx, ws_lognorm,
                                                          out_ll);
}